// DWTLayer_24532853195398
// MI455X (gfx1250) — compile-verified
//
#include <hip/hip_runtime.h>
#include <hip/hip_bf16.h>
#include <stdint.h>

typedef float v4f __attribute__((ext_vector_type(4)));

// x: [16,256,256,64] f32, NHWC, channels contiguous.
// out: 4 concatenated [16,128,128,64] f32 tensors (cA,cH,cV,cD).
// Pure streaming kernel (every byte touched exactly once):
//   - stage 2 contiguous 16KB input-row spans per block into LDS via
//     global_load_async_to_lds_b128 (ASYNCcnt path), non-temporal,
//   - s_wait_asynccnt 0 + barrier,
//   - Haar butterfly from LDS (ds_load_b128),
//   - 4 coalesced non-temporal B128 store streams.

#define C_CH        64
#define W_IN        256
#define ROW_FLOATS  (W_IN * C_CH)        // 16384 floats per input row
#define SEG_FLOATS  (64 * C_CH)          // 4096 floats per staged row segment
#define SEG_CHUNKS  (SEG_FLOATS / 4)     // 1024 x 16B chunks per row
#define OUT_PLANE_F4 (16u * 128u * 128u * 64u / 4u)  // 4,194,304 float4 per output tensor

__global__ __launch_bounds__(256) void DWTLayer_24532853195398_kernel(
    const float* __restrict__ x, float* __restrict__ out) {
    __shared__ v4f sh[2 * SEG_CHUNKS];   // 32 KB: row0 then row1

    const int tid = threadIdx.x;
    const unsigned bid = blockIdx.x;       // 8192 blocks = 16*128*4
    const int seg = bid & 3;               // which 32-wide output-w segment
    const int h   = (bid >> 2) & 127;      // output row
    const int n   = bid >> 9;              // batch

    // Input base offsets (in floats) for the two staged row segments.
    const size_t row0 = ((size_t)(n * 256 + 2 * h) * W_IN + (size_t)seg * 64) * C_CH;
    const size_t row1 = row0 + ROW_FLOATS;

    const float* base0 = x + row0;         // uniform -> SGPR pair
    const float* base1 = x + row1;
    const uint32_t ldsbase = (uint32_t)(uintptr_t)(void*)&sh[0];

    // Async stage: 2 rows x 1024 chunks of 16B; 256 threads -> 4 chunks each per row.
    // Non-temporal: input is streamed exactly once, keep it out of L2 residency.
    #pragma unroll
    for (int i = 0; i < 4; ++i) {
        const uint32_t chunk = (uint32_t)tid + (uint32_t)i * 256u;  // 0..1023
        const uint32_t goff  = chunk * 16u;                          // byte offset in row span
        const uint32_t l0    = ldsbase + chunk * 16u;
        const uint32_t l1    = ldsbase + (uint32_t)(SEG_CHUNKS * 16) + chunk * 16u;
        asm volatile("global_load_async_to_lds_b128 %0, %1, %2 th:TH_LOAD_NT"
                     :: "v"(l0), "v"(goff), "s"(base0) : "memory");
        asm volatile("global_load_async_to_lds_b128 %0, %1, %2 th:TH_LOAD_NT"
                     :: "v"(l1), "v"(goff), "s"(base1) : "memory");
    }
    // Each wave waits for its own async copies to land in LDS, then barrier
    // makes all waves' staged data visible workgroup-wide.
    asm volatile("s_wait_asynccnt 0" ::: "memory");
    __syncthreads();

    v4f* __restrict__ o4 = (v4f*)out;

    // 32 output pixels * 16 float4-of-channels = 512 items; 2 per thread.
    #pragma unroll
    for (int j = tid; j < 512; j += 256) {
        const int c4 = j & 15;       // float4 index within the 64 channels
        const int wl = j >> 4;       // local output w in [0,32)

        const int p0 = (2 * wl) * 16 + c4;       // float4 index of input pixel 2w
        const v4f a = sh[p0];                     // row0, 2w
        const v4f b = sh[p0 + 16];                // row0, 2w+1
        const v4f c = sh[SEG_CHUNKS + p0];        // row1, 2w
        const v4f d = sh[SEG_CHUNKS + p0 + 16];   // row1, 2w+1

        const v4f sAB = a + b, dAB = a - b;
        const v4f sCD = c + d, dCD = c - d;
        const v4f cA = 0.5f * (sAB + sCD);
        const v4f cH = 0.5f * (dAB + dCD);
        const v4f cV = 0.5f * (sAB - sCD);
        const v4f cD = -cH;                       // -HIGH filter == negated HIGH

        const int wg = seg * 32 + wl;
        const size_t o = ((size_t)((n * 128 + h) * 128 + wg)) * 16 + (size_t)c4;
        __builtin_nontemporal_store(cA, &o4[o]);
        __builtin_nontemporal_store(cH, &o4[o + OUT_PLANE_F4]);
        __builtin_nontemporal_store(cV, &o4[o + 2 * OUT_PLANE_F4]);
        __builtin_nontemporal_store(cD, &o4[o + 3 * OUT_PLANE_F4]);
    }
}

extern "C" void kernel_launch(void* const* d_in, const int* in_sizes, int n_in,
                              void* d_out, int out_size, void* d_ws, size_t ws_size,
                              hipStream_t stream) {
    (void)in_sizes; (void)n_in; (void)out_size; (void)d_ws; (void)ws_size;
    const float* x = (const float*)d_in[0];
    float* out = (float*)d_out;
    // 16 (n) * 128 (h) * 4 (w segments) = 8192 blocks, 256 threads each.
    DWTLayer_24532853195398_kernel<<<dim3(8192), dim3(256), 0, stream>>>(x, out);
}